// TemporalLoss_5050881540163
// MI455X (gfx1250) — compile-verified
//
#include <hip/hip_runtime.h>
#include <math.h>

// CDNA5 / gfx1250 — wave32, WMMA (not MFMA).
typedef float v2f __attribute__((ext_vector_type(2)));
typedef float v8f __attribute__((ext_vector_type(8)));

static constexpr int B_ROWS  = 4096;
static constexpr int T_COLS  = 8192;
static constexpr int THREADS = 256;        // 8 wave32 waves per workgroup

// ---------------------------------------------------------------------------
// Kernel 1: per-row raw moments  (Sp, St, Spp, Stt, Spt)
// One workgroup per row. Streaming float4 loads (global_load_b128), fp32 FMA
// accumulation in registers, then a WMMA-based tree reduction of the 256
// per-thread partials using V_WMMA_F32_16X16X4_F32 with an all-ones B matrix
// (D[m][n] = sum_k A[m][k], so each WMMA row-sums a 16x4 tile of partials).
// ---------------------------------------------------------------------------
__global__ __launch_bounds__(THREADS)
void temporal_row_stats_kernel(const float* __restrict__ pred,
                               const float* __restrict__ target,
                               float* __restrict__ stats /* [B_ROWS][5] */) {
    const int row = blockIdx.x;
    const int tid = threadIdx.x;

    const float4* __restrict__ p4 =
        reinterpret_cast<const float4*>(pred + (size_t)row * T_COLS);
    const float4* __restrict__ t4 =
        reinterpret_cast<const float4*>(target + (size_t)row * T_COLS);

    constexpr int VEC_PER_ROW = T_COLS / 4;          // 2048 float4 per row
    constexpr int ITERS       = VEC_PER_ROW / THREADS; // 8

    float sp = 0.f, st = 0.f, spp = 0.f, stt = 0.f, spt = 0.f;

#pragma unroll
    for (int i = 0; i < ITERS; ++i) {
        const int idx = i * THREADS + tid;           // fully coalesced b128
        const float4 p = p4[idx];
        const float4 t = t4[idx];

        sp += (p.x + p.y) + (p.z + p.w);
        st += (t.x + t.y) + (t.z + t.w);

        spp = fmaf(p.x, p.x, spp); spp = fmaf(p.y, p.y, spp);
        spp = fmaf(p.z, p.z, spp); spp = fmaf(p.w, p.w, spp);

        stt = fmaf(t.x, t.x, stt); stt = fmaf(t.y, t.y, stt);
        stt = fmaf(t.z, t.z, stt); stt = fmaf(t.w, t.w, stt);

        spt = fmaf(p.x, t.x, spt); spt = fmaf(p.y, t.y, spt);
        spt = fmaf(p.z, t.z, spt); spt = fmaf(p.w, t.w, spt);
    }

    // Stage the 5 per-thread partials into LDS: smem[q*256 + tid].
    __shared__ float smem[5 * THREADS];
    smem[0 * THREADS + tid] = sp;
    smem[1 * THREADS + tid] = st;
    smem[2 * THREADS + tid] = spp;
    smem[3 * THREADS + tid] = stt;
    smem[4 * THREADS + tid] = spt;
    __syncthreads();

    // Wave 0 (uniform branch, EXEC all ones within the wave) reduces each set
    // of 256 partials with 4 accumulating WMMAs + a half-wave fold.
    if (tid < 32) {
        const int  lane = tid;
        const bool hi   = lane >= 16;
        const int  m    = hi ? lane - 16 : lane;     // A-matrix row (M)
        const int  kOff = hi ? 2 : 0;                // K=0/1 low half, K=2/3 high

        v2f ones;
        ones[0] = 1.0f;
        ones[1] = 1.0f;                              // B = 4x16 all-ones

        float tot[5];
#pragma unroll
        for (int q = 0; q < 5; ++q) {
            v8f acc = {};
#pragma unroll
            for (int tile = 0; tile < 4; ++tile) {
                // A[m][k] = partial[tile*64 + m*4 + k]
                const int base = q * THREADS + tile * 64 + m * 4 + kOff;
                v2f a;
                a[0] = smem[base];                   // K = kOff
                a[1] = smem[base + 1];               // K = kOff + 1
                // D = A x Ones + C  -> each row of D accumulates its 4 partials
                acc = __builtin_amdgcn_wmma_f32_16x16x4_f32(
                        /*neg_a=*/false, a,
                        /*neg_b=*/false, ones,
                        /*c_mod=*/(short)0, acc,
                        /*reuse_a=*/false, /*reuse_b=*/false);
            }
            // acc[j]: lanes 0-15 hold row M=j, lanes 16-31 hold row M=j+8
            // (columns identical under ones-B). Sum rows, then fold halves.
            float s = ((acc[0] + acc[1]) + (acc[2] + acc[3])) +
                      ((acc[4] + acc[5]) + (acc[6] + acc[7]));
            s += __shfl_xor(s, 16, 32);              // combine M 0-7 with 8-15
            tot[q] = s;
        }

        if (lane == 0) {
#pragma unroll
            for (int q = 0; q < 5; ++q)
                stats[(size_t)row * 5 + q] = tot[q];
        }
    }
}

// ---------------------------------------------------------------------------
// Kernel 2: fold 4096 row-stat records into the final scalar:
//   out = mean((p-t)^2) / mean(|corr_row|)
// ---------------------------------------------------------------------------
__global__ __launch_bounds__(THREADS)
void temporal_finalize_kernel(const float* __restrict__ stats,
                              float* __restrict__ out) {
    const int tid = threadIdx.x;
    const float invT = 1.0f / (float)T_COLS;

    float msesum = 0.f, corrsum = 0.f;
    for (int b = tid; b < B_ROWS; b += THREADS) {
        const float sp  = stats[b * 5 + 0];
        const float st  = stats[b * 5 + 1];
        const float spp = stats[b * 5 + 2];
        const float stt = stats[b * 5 + 3];
        const float spt = stats[b * 5 + 4];

        msesum += (spp - 2.0f * spt) + stt;          // sum (p-t)^2 for this row

        const float num  = spt - sp * st * invT;     // centered cross-moment
        const float varp = spp - sp * sp * invT;
        const float vart = stt - st * st * invT;
        corrsum += fabsf(num / sqrtf(varp * vart));
    }

    __shared__ float sm[2 * THREADS];
    sm[tid]           = msesum;
    sm[THREADS + tid] = corrsum;
    __syncthreads();

#pragma unroll
    for (int s = THREADS / 2; s > 0; s >>= 1) {
        if (tid < s) {
            sm[tid]           += sm[tid + s];
            sm[THREADS + tid] += sm[THREADS + tid + s];
        }
        __syncthreads();
    }

    if (tid == 0) {
        const float mse      = sm[0] / ((float)B_ROWS * (float)T_COLS);
        const float meancorr = sm[THREADS] / (float)B_ROWS;
        out[0] = mse / meancorr;
    }
}

// ---------------------------------------------------------------------------
extern "C" void kernel_launch(void* const* d_in, const int* in_sizes, int n_in,
                              void* d_out, int out_size, void* d_ws, size_t ws_size,
                              hipStream_t stream) {
    (void)in_sizes; (void)n_in; (void)out_size; (void)ws_size;

    const float* pred   = (const float*)d_in[0];
    const float* target = (const float*)d_in[1];
    float*       outp   = (float*)d_out;
    float*       stats  = (float*)d_ws;   // needs 4096*5*4 = 80 KB scratch

    temporal_row_stats_kernel<<<B_ROWS, THREADS, 0, stream>>>(pred, target, stats);
    temporal_finalize_kernel<<<1, THREADS, 0, stream>>>(stats, outp);
}